// SVDBottleneck_33517924778319
// MI455X (gfx1250) — compile-verified
//
#include <hip/hip_runtime.h>

typedef float v2f __attribute__((ext_vector_type(2)));
typedef float v8f __attribute__((ext_vector_type(8)));

#define NROW  256          // matrix is 256 x 256
#define KDIM  64           // subspace width (>= RANK, multiple of 16)
#define RANK  50
#define LDQ   68           // padded leading dim for LDS panels (bank-conflict offset)
#define NITER 8            // subspace iterations

__device__ __forceinline__ v8f wmma_f32(v2f a, v2f b, v8f c) {
    // D = A(16x4) * B(4x16) + C(16x16), fp32 everywhere
    return __builtin_amdgcn_wmma_f32_16x16x4_f32(
        /*neg_a=*/false, a, /*neg_b=*/false, b,
        /*c_mod=*/(short)0, c, /*reuse_a=*/false, /*reuse_b=*/false);
}

// C[256 x KDIM] (LDS, ld=LDQ) = op(A)[256x256] (global) * B[256 x KDIM] (LDS, ld=LDQ)
// TRANS=false: op(A)=A ; TRANS=true: op(A)=A^T
template <bool TRANS>
__device__ void gemm_A_x(const float* __restrict__ A, const float* B, float* C,
                         int wave, int lh, int ll) {
    for (int itb = wave; itb < 16; itb += 8) {       // 16 row-blocks of 16
        v8f acc[4] = {};
        for (int k0 = 0; k0 < NROW; k0 += 4) {
            v2f a;
            if (!TRANS) {
                // A-frag: lane ll holds row m, VGPR pair holds K = k0 + 2*lh + {0,1}
                int row = itb * 16 + ll;
                int kk  = k0 + 2 * lh;
                a.x = A[row * NROW + kk];
                a.y = A[row * NROW + kk + 1];
            } else {
                // (A^T)[m, kk] = A[kk, m] ; loads coalesced across lanes
                int col = itb * 16 + ll;
                int kk  = k0 + 2 * lh;
                a.x = A[kk * NROW + col];
                a.y = A[(kk + 1) * NROW + col];
            }
            for (int jt = 0; jt < 4; ++jt) {
                v2f b;
                int nn = jt * 16 + ll;
                b.x = B[(k0 + 2 * lh) * LDQ + nn];
                b.y = B[(k0 + 2 * lh + 1) * LDQ + nn];
                acc[jt] = wmma_f32(a, b, acc[jt]);
            }
        }
        for (int jt = 0; jt < 4; ++jt)
            for (int v = 0; v < 8; ++v) {
                int m  = itb * 16 + v + 8 * lh;      // C/D layout: row = vgpr + 8*half
                int nn = jt * 16 + ll;
                C[m * LDQ + nn] = acc[jt][v];
            }
    }
}

// S[KDIM x KDIM] (ld=LDQ) = Y^T * Y, Y is 256 x KDIM in LDS (ld=LDQ)
__device__ void gemm_gram(const float* Y, float* S, int wave, int lh, int ll) {
    for (int t = wave; t < 16; t += 8) {             // 4x4 tile grid
        int it = t >> 2, jt = t & 3;
        v8f acc = {};
        for (int k0 = 0; k0 < NROW; k0 += 4) {
            v2f a, b;
            a.x = Y[(k0 + 2 * lh) * LDQ + it * 16 + ll];       // (Y^T)[m,kk]=Y[kk,m]
            a.y = Y[(k0 + 2 * lh + 1) * LDQ + it * 16 + ll];
            b.x = Y[(k0 + 2 * lh) * LDQ + jt * 16 + ll];
            b.y = Y[(k0 + 2 * lh + 1) * LDQ + jt * 16 + ll];
            acc = wmma_f32(a, b, acc);
        }
        for (int v = 0; v < 8; ++v)
            S[(it * 16 + v + 8 * lh) * LDQ + jt * 16 + ll] = acc[v];
    }
}

__global__ void __launch_bounds__(256)
svd_rank50_kernel(const float* __restrict__ X, float* __restrict__ Out) {
    extern __shared__ float smem[];
    float* Q = smem;                       // 256 x KDIM  (also holds Y)
    float* T = smem + NROW * LDQ;          // 256 x KDIM
    float* S = smem + 2 * NROW * LDQ;      // KDIM x KDIM Gram / Cholesky factor

    const int n = blockIdx.x;
    const float* A = X + (size_t)n * NROW * NROW;
    float* O = Out + (size_t)n * NROW * NROW;

    const int tid  = threadIdx.x;
    const int wave = tid >> 5;
    const int lane = tid & 31;
    const int lh   = lane >> 4;   // which 16-lane half
    const int ll   = lane & 15;

    // ---- init Q0 with a deterministic pseudo-random hash ----
    for (int idx = tid; idx < NROW * KDIM; idx += 256) {
        int r = idx / KDIM, c = idx % KDIM;
        unsigned h = (unsigned)(r * 131 + c * 2654435761u) ^ 0x9e3779b9u;
        h ^= h >> 16; h *= 0x85ebca6bu; h ^= h >> 13;
        Q[r * LDQ + c] = ((float)(h & 0xffff) * (1.0f / 32768.0f)) - 1.0f;
    }
    __syncthreads();

    // ---- subspace iteration: Y = A^T (A Q); Q = cholQR(Y) ----
    for (int iter = 0; iter < NITER; ++iter) {
        gemm_A_x<false>(A, Q, T, wave, lh, ll);   // T = A * Q
        __syncthreads();
        gemm_A_x<true>(A, T, Q, wave, lh, ll);    // Y = A^T * T  (into Q buffer)
        __syncthreads();
        gemm_gram(Q, S, wave, lh, ll);            // S = Y^T Y
        __syncthreads();

        // Cholesky S = L L^T (lower triangle of S), parallel right-looking
        for (int k = 0; k < KDIM; ++k) {
            if (tid == 0) S[k * LDQ + k] = sqrtf(fmaxf(S[k * LDQ + k], 1e-30f));
            __syncthreads();
            if (tid > k && tid < KDIM) S[tid * LDQ + k] /= S[k * LDQ + k];
            __syncthreads();
            for (int idx = tid; idx < KDIM * KDIM; idx += 256) {
                int i = idx >> 6, j = idx & 63;
                if (i > k && j > k && j <= i)
                    S[i * LDQ + j] -= S[i * LDQ + k] * S[j * LDQ + k];
            }
            __syncthreads();
        }

        // Q = Y * L^{-T}: per-row forward substitution, one row per thread, in place
        {
            float* row = Q + tid * LDQ;
            for (int j = 0; j < KDIM; ++j) {
                float v = row[j];
                for (int i = 0; i < j; ++i) v -= row[i] * S[j * LDQ + i];
                row[j] = v / S[j * LDQ + j];
            }
        }
        __syncthreads();
    }

    // ---- keep leading RANK columns only ----
    for (int idx = tid; idx < NROW * (KDIM - RANK); idx += 256) {
        int r = idx / (KDIM - RANK);
        int c = RANK + idx % (KDIM - RANK);
        Q[r * LDQ + c] = 0.0f;
    }
    __syncthreads();

    gemm_A_x<false>(A, Q, T, wave, lh, ll);       // T = A * Q_50
    __syncthreads();

    // ---- Out = T * Q^T : 256x256 = (256xKDIM)*(KDIMx256) ----
    for (int t = wave; t < 256; t += 8) {         // 16x16 tile grid, wave-uniform
        int it = t >> 4, jt = t & 15;
        v8f acc = {};
        for (int k0 = 0; k0 < KDIM; k0 += 4) {
            v2f a, b;
            a.x = T[(it * 16 + ll) * LDQ + k0 + 2 * lh];
            a.y = T[(it * 16 + ll) * LDQ + k0 + 2 * lh + 1];
            // B[k,n] = Q^T[k,n] = Q[n,k]
            b.x = Q[(jt * 16 + ll) * LDQ + k0 + 2 * lh];
            b.y = Q[(jt * 16 + ll) * LDQ + k0 + 2 * lh + 1];
            acc = wmma_f32(a, b, acc);
        }
        for (int v = 0; v < 8; ++v)
            O[(it * 16 + v + 8 * lh) * NROW + jt * 16 + ll] = acc[v];
    }
}

extern "C" void kernel_launch(void* const* d_in, const int* in_sizes, int n_in,
                              void* d_out, int out_size, void* d_ws, size_t ws_size,
                              hipStream_t stream) {
    const float* x = (const float*)d_in[0];
    float* out = (float*)d_out;
    const int nmat = in_sizes[0] / (NROW * NROW);   // 64*3 = 192
    const size_t shmem = (size_t)(2 * NROW * LDQ + KDIM * LDQ) * sizeof(float);
    (void)hipFuncSetAttribute((const void*)svd_rank50_kernel,
                              hipFuncAttributeMaxDynamicSharedMemorySize,
                              (int)shmem);
    svd_rank50_kernel<<<nmat, 256, shmem, stream>>>(x, out);
}